// EnhancedMultiTaskLoss_8701603742268
// MI455X (gfx1250) — compile-verified
//
#include <hip/hip_runtime.h>

// EnhancedMultiTaskLoss for MI455X (gfx1250, wave32).
// Bandwidth-bound streaming reduction: ~168 MB read -> ~7.2 us at 23.3 TB/s.
// Pass 1: 2048 blocks x 256 threads, pair-vectorized b128/b64 NT loads,
//         per-lane accumulation, wave reduction via V_WMMA_F32_16X16X4_F32
//         (A = (acc,0) per lane, B = ones -> row sums = 2-lane partials),
//         per-block partials to d_ws (deterministic, no atomics).
// Pass 2: single block reduces 2048 partial pairs in a fixed tree, writes
//         (total, cls_loss, vad_loss, micl_loss).

typedef __attribute__((ext_vector_type(2))) float v2f;
typedef __attribute__((ext_vector_type(4))) float v4f;
typedef __attribute__((ext_vector_type(8))) float v8f;
typedef __attribute__((ext_vector_type(2))) int   v2i;

#define NBLK  2048
#define NTHR  256
#define NWAVE (NTHR / 32)

__constant__ float VAD_TAB[6][3] = {
    { 0.96f,   0.648f,  0.588f},
    {-0.896f, -0.424f, -0.672f},
    {-0.062f, -0.632f, -0.286f},
    {-0.666f,  0.73f,   0.314f},
    { 0.85f,   0.75f,   0.45f },
    {-0.5f,    0.6f,   -0.2f  },
};

// Full 32-lane f32 sum using one V_WMMA_F32_16X16X4_F32.
// A(16x4) layout: lane L(0-15): V0=A[L][0], V1=A[L][1]; lane L+16: V0=A[L][2], V1=A[L][3].
// With A = (acc, 0) and B = ones(4x16) (layout-independent), D[m][n] = acc(m)+acc(m+16).
// Per lane, sum of the 8 D VGPRs covers rows 0-7 (lanes<16) or 8-15 (lanes>=16);
// one shfl_xor(16) completes the wave sum. EXEC is all-ones at call sites.
__device__ __forceinline__ float wave_sum_wmma(float acc) {
    v2f a; a[0] = acc;  a[1] = 0.0f;
    v2f b; b[0] = 1.0f; b[1] = 1.0f;
    v8f c = {};
    v8f d = __builtin_amdgcn_wmma_f32_16x16x4_f32(
        /*neg_a=*/false, a, /*neg_b=*/false, b,
        /*c_mod=*/(short)0, c, /*reuse_a=*/false, /*reuse_b=*/false);
    float s = d[0] + d[1] + d[2] + d[3] + d[4] + d[5] + d[6] + d[7];
    s += __shfl_xor(s, 16, 32);
    return s;
}

__device__ __forceinline__ void accum_sample(const float* x, int lab, const float* vp,
                                             float& facc, float& vacc) {
    float mx = fmaxf(fmaxf(fmaxf(x[0], x[1]), fmaxf(x[2], x[3])), fmaxf(x[4], x[5]));
    float e[6];
    float s = 0.0f;
#pragma unroll
    for (int c = 0; c < 6; ++c) { e[c] = __expf(x[c] - mx); s += e[c]; }
    float invs = 1.0f / s;
    float lse  = __logf(s);
    float fsum = 0.0f;
#pragma unroll
    for (int c = 0; c < 6; ++c) {
        float logp = (x[c] - mx) - lse;
        float p    = e[c] * invs;
        float om   = 1.0f - p;
        float t    = (c == lab) ? 0.9f : 0.02f;   // 1-eps / eps/(C-1)
        fsum = fmaf(om * om * t, logp, fsum);
    }
    facc -= fsum;                                  // focal = -sum(...)
    float d0 = vp[0] - VAD_TAB[lab][0];
    float d1 = vp[1] - VAD_TAB[lab][1];
    float d2 = vp[2] - VAD_TAB[lab][2];
    vacc = fmaf(d0, d0, vacc);
    vacc = fmaf(d1, d1, vacc);
    vacc = fmaf(d2, d2, vacc);
}

__global__ __launch_bounds__(NTHR)
void focal_vad_partial(const float* __restrict__ logits,
                       const int*   __restrict__ labels,
                       const float* __restrict__ vad_pred,
                       float*       __restrict__ partials,
                       int n) {
    float facc = 0.0f, vacc = 0.0f;
    const int npairs = n >> 1;
    const int stride = gridDim.x * blockDim.x;
    for (int p = blockIdx.x * blockDim.x + threadIdx.x; p < npairs; p += stride) {
        // two samples per lane: 48B logits (3 x b128), 24B vad (3 x b64), 8B labels
        const float* lp = logits + (size_t)p * 12;
        v4f l0 = __builtin_nontemporal_load((const v4f*)(lp + 0));
        v4f l1 = __builtin_nontemporal_load((const v4f*)(lp + 4));
        v4f l2 = __builtin_nontemporal_load((const v4f*)(lp + 8));
        const float* vp = vad_pred + (size_t)p * 6;
        v2f w0 = __builtin_nontemporal_load((const v2f*)(vp + 0));
        v2f w1 = __builtin_nontemporal_load((const v2f*)(vp + 2));
        v2f w2 = __builtin_nontemporal_load((const v2f*)(vp + 4));
        v2i lb = __builtin_nontemporal_load((const v2i*)(labels + (size_t)p * 2));

        float xa[6] = {l0[0], l0[1], l0[2], l0[3], l1[0], l1[1]};
        float va[3] = {w0[0], w0[1], w1[0]};
        accum_sample(xa, lb[0], va, facc, vacc);

        float xb[6] = {l1[2], l1[3], l2[0], l2[1], l2[2], l2[3]};
        float vb[3] = {w1[1], w2[0], w2[1]};
        accum_sample(xb, lb[1], vb, facc, vacc);
    }
    // odd-n tail (not hit for B = 4M, kept for generality)
    if ((n & 1) && blockIdx.x == 0 && threadIdx.x == 0) {
        int i = n - 1;
        const float* lp = logits + (size_t)i * 6;
        float xt[6] = {lp[0], lp[1], lp[2], lp[3], lp[4], lp[5]};
        const float* vp = vad_pred + (size_t)i * 3;
        float vt[3] = {vp[0], vp[1], vp[2]};
        accum_sample(xt, labels[i], vt, facc, vacc);
    }

    float wf = wave_sum_wmma(facc);
    float wv = wave_sum_wmma(vacc);

    __shared__ float lf[NWAVE], lv[NWAVE];
    int wave = threadIdx.x >> 5;
    int lane = threadIdx.x & 31;
    if (lane == 0) { lf[wave] = wf; lv[wave] = wv; }
    __syncthreads();
    if (threadIdx.x == 0) {
        float F = 0.0f, V = 0.0f;
#pragma unroll
        for (int w = 0; w < NWAVE; ++w) { F += lf[w]; V += lv[w]; }
        partials[2 * blockIdx.x]     = F;
        partials[2 * blockIdx.x + 1] = V;
    }
}

__global__ __launch_bounds__(NTHR)
void focal_vad_final(const float* __restrict__ partials,
                     const float* __restrict__ micl,
                     float*       __restrict__ out,
                     int nparts, float invB, float inv3B) {
    float F = 0.0f, V = 0.0f;
    for (int j = threadIdx.x; j < nparts; j += NTHR) {   // fixed order: deterministic
        F += partials[2 * j];
        V += partials[2 * j + 1];
    }
    F = wave_sum_wmma(F);
    V = wave_sum_wmma(V);
    __shared__ float lf[NWAVE], lv[NWAVE];
    int wave = threadIdx.x >> 5;
    int lane = threadIdx.x & 31;
    if (lane == 0) { lf[wave] = F; lv[wave] = V; }
    __syncthreads();
    if (threadIdx.x == 0) {
        float Ft = 0.0f, Vt = 0.0f;
#pragma unroll
        for (int w = 0; w < NWAVE; ++w) { Ft += lf[w]; Vt += lv[w]; }
        float cls = Ft * invB;
        float vad = Vt * inv3B;
        float m   = micl[0];
        out[0] = cls + 0.5f * vad + 0.3f * m;   // total
        out[1] = cls;
        out[2] = vad;
        out[3] = m;
    }
}

extern "C" void kernel_launch(void* const* d_in, const int* in_sizes, int n_in,
                              void* d_out, int out_size, void* d_ws, size_t ws_size,
                              hipStream_t stream) {
    (void)n_in; (void)out_size; (void)ws_size;
    const float* logits   = (const float*)d_in[0];
    const int*   labels   = (const int*)  d_in[1];
    const float* vad_pred = (const float*)d_in[2];
    const float* micl     = (const float*)d_in[3];
    float* out      = (float*)d_out;
    float* partials = (float*)d_ws;      // 2 * NBLK floats = 16 KB scratch

    const int n = in_sizes[1];           // B (labels element count)
    focal_vad_partial<<<NBLK, NTHR, 0, stream>>>(logits, labels, vad_pred, partials, n);
    const float invB  = 1.0f / (float)n;
    const float inv3B = invB / 3.0f;
    focal_vad_final<<<1, NTHR, 0, stream>>>(partials, micl, out, NBLK, invB, inv3B);
}